// EEGNetGNN_74801150427322
// MI455X (gfx1250) — compile-verified
//
#include <hip/hip_runtime.h>
#include <hip/hip_bf16.h>

#define B_  16
#define T_  1536
#define V_  64
#define F1_ 16
#define F2_ 32
#define H_  4
#define FO_ 8
#define K1_ 32
#define K3_ 16
#define E_  320
#define TT_ 16
#define EPS_ 1e-5f
#define NEG_SLOPE_ 0.2f

typedef float v2f __attribute__((ext_vector_type(2)));
typedef float v8f __attribute__((ext_vector_type(8)));

__device__ __forceinline__ v8f wmma_f32_k4(v2f a, v2f b, v8f c) {
  // D = A(16x4) * B(4x16) + C, full f32 precision
  return __builtin_amdgcn_wmma_f32_16x16x4_f32(false, a, false, b, (short)0, c,
                                               false, false);
}

// ---------------------------------------------------------------------------
// Kernel 0: fold conv1 + bn1 + GAT linear into Weff[k][ho] (32x32) and cb[32]
//   Weff[k,ho] = sum_f a1[f] * w1[f,k] * W[f,ho];  cb[ho] = sum_f c1[f]*W[f,ho]
// ---------------------------------------------------------------------------
__global__ void k0_fold(const float* __restrict__ w1, const float* __restrict__ g1,
                        const float* __restrict__ b1, const float* __restrict__ m1,
                        const float* __restrict__ v1, const float* __restrict__ gw,
                        float* __restrict__ wsW, float* __restrict__ wsCb) {
  __shared__ float a1[F1_], c1[F1_];
  int tid = threadIdx.x;
  if (tid < F1_) {
    float a = g1[tid] * rsqrtf(v1[tid] + EPS_);
    a1[tid] = a;
    c1[tid] = b1[tid] - a * m1[tid];
  }
  __syncthreads();
  for (int idx = tid; idx < K1_ * F2_; idx += blockDim.x) {
    int k = idx >> 5, ho = idx & 31;
    float s = 0.f;
#pragma unroll
    for (int f = 0; f < F1_; ++f)
      s += a1[f] * w1[f * K1_ + k] * gw[f * F2_ + ho];
    wsW[idx] = s;
  }
  if (tid < F2_) {
    float s = 0.f;
#pragma unroll
    for (int f = 0; f < F1_; ++f) s += c1[f] * gw[f * F2_ + tid];
    wsCb[tid] = s;
  }
}

// ---------------------------------------------------------------------------
// Kernel 1: fused conv1+bn1+GAT-linear (WMMA) + attention + mean over V
//   One block = (b, 16 consecutive t).  8 waves: 2 ho-tiles x 4 v-tiles.
//   A-fragments (folded weights) are hoisted into registers across the t-loop;
//   only the B (input-window) fragments are re-read from LDS per t.
// ---------------------------------------------------------------------------
__global__ __launch_bounds__(256) void k1_gat(
    const float* __restrict__ x, const float* __restrict__ asrc,
    const float* __restrict__ adst, const float* __restrict__ gbias,
    const int* __restrict__ edge, const float* __restrict__ wsW,
    const float* __restrict__ wsCb, float* __restrict__ gbuf) {
  __shared__ float xs[V_][48];        // input windows: t0-15 .. t0+31
  __shared__ float wf[K1_][F2_];      // Weff[k][ho]
  __shared__ float cbs[F2_];
  __shared__ float asL[H_][FO_], adL[H_][FO_];
  __shared__ float gbL[F2_];
  __shared__ int   srcL[E_], dstL[E_];
  __shared__ float yv[V_][F2_ + 1];   // y for current t (padded)
  __shared__ float esL[V_][H_], edL[V_][H_];
  __shared__ float denL[V_][H_];
  __shared__ float exL[E_][H_];
  __shared__ float outL[V_][F2_];

  const int tid = threadIdx.x;
  const int b = blockIdx.y;
  const int t0 = blockIdx.x * TT_;

  for (int i = tid; i < K1_ * F2_; i += 256) wf[i >> 5][i & 31] = wsW[i];
  if (tid < F2_) { cbs[tid] = wsCb[tid]; gbL[tid] = gbias[tid]; }
  if (tid < H_ * FO_) {
    asL[tid >> 3][tid & 7] = asrc[tid];
    adL[tid >> 3][tid & 7] = adst[tid];
  }
  for (int e = tid; e < E_; e += 256) { srcL[e] = edge[e]; dstL[e] = edge[E_ + e]; }
  for (int i = tid; i < V_ * 47; i += 256) {
    int v = i / 47, col = i % 47;
    int tg = t0 + col - 15;  // conv 'same' left pad = 15
    xs[v][col] = (tg >= 0 && tg < T_) ? x[(b * V_ + v) * T_ + tg] : 0.f;
  }
  __syncthreads();

  const int lane  = tid & 31;
  const int wv    = tid >> 5;   // wave 0..7
  const int hoT   = wv & 1;     // ho tile (0: ho 0-15, 1: ho 16-31)
  const int vT    = wv >> 1;    // v tile 0..3
  const int ln15  = lane & 15;
  const int lhalf = lane >> 4;  // selects K pair per ISA A/B layout

  // Hoist all A fragments (Weff^T tiles) into registers: invariant over tl.
  v2f aW[8];
#pragma unroll
  for (int kb = 0; kb < 8; ++kb) {
    int k0 = kb * 4 + 2 * lhalf;
    aW[kb].x = wf[k0][hoT * 16 + ln15];      // A[m=ho][k]
    aW[kb].y = wf[k0 + 1][hoT * 16 + ln15];
  }

  for (int tl = 0; tl < TT_; ++tl) {
    denL[tid >> 2][tid & 3] = 0.f;
    for (int i = tid; i < V_ * F2_; i += 256) outL[i >> 5][i & 31] = 0.f;

    // ---- WMMA: D[ho][v] = Weff^T x Xwin, K=32 in 8 steps of 4 ----
    v8f acc = {0.f, 0.f, 0.f, 0.f, 0.f, 0.f, 0.f, 0.f};
#pragma unroll
    for (int kb = 0; kb < 8; ++kb) {
      int k0 = kb * 4 + 2 * lhalf;
      v2f bm;
      bm.x = xs[vT * 16 + ln15][tl + k0];    // B[k][n=v]
      bm.y = xs[vT * 16 + ln15][tl + k0 + 1];
      acc = wmma_f32_k4(aW[kb], bm, acc);
    }
    {
      int v = vT * 16 + ln15;
#pragma unroll
      for (int i = 0; i < 8; ++i) {
        int ho = hoT * 16 + i + 8 * lhalf;   // D layout: M = i + 8*(lane>=16)
        yv[v][ho] = acc[i] + cbs[ho];
      }
    }
    __syncthreads();

    // ---- attention scores per node/head ----
    {
      int v = tid >> 2, h = tid & 3;
      float s = 0.f, d = 0.f;
#pragma unroll
      for (int o = 0; o < FO_; ++o) {
        float yy = yv[v][h * FO_ + o];
        s += yy * asL[h][o];
        d += yy * adL[h][o];
      }
      esL[v][h] = s;
      edL[v][h] = d;
    }
    __syncthreads();

    // ---- pass A: exp(leaky_relu) + segment denominator ----
    for (int idx = tid; idx < E_ * H_; idx += 256) {
      int e = idx >> 2, h = idx & 3;
      float sc = esL[srcL[e]][h] + edL[dstL[e]][h];
      sc = sc > 0.f ? sc : NEG_SLOPE_ * sc;
      float ex = __expf(sc);
      exL[e][h] = ex;
      atomicAdd(&denL[dstL[e]][h], ex);
    }
    __syncthreads();

    // ---- pass B: out[dst] += alpha * y[src] ----
    for (int idx = tid; idx < E_ * H_; idx += 256) {
      int e = idx >> 2, h = idx & 3;
      int s = srcL[e], d = dstL[e];
      float al = exL[e][h] / denL[d][h];
#pragma unroll
      for (int o = 0; o < FO_; ++o)
        atomicAdd(&outL[d][h * FO_ + o], al * yv[s][h * FO_ + o]);
    }
    __syncthreads();

    // ---- mean over V + bias -> gbuf[b][ho][t] (B,F2,T layout) ----
    if (tid < F2_) {
      float s = 0.f;
      for (int v = 0; v < V_; ++v) s += outL[v][tid];
      gbuf[(b * F2_ + tid) * T_ + (t0 + tl)] = s * (1.f / V_) + gbL[tid];
    }
    __syncthreads();
  }
}

// ---------------------------------------------------------------------------
// Kernel 2: bn2 + ELU + avgpool8  -> p1[b][c][192]
// ---------------------------------------------------------------------------
__global__ void k2_pool8(const float* __restrict__ gbuf,
                         const float* __restrict__ g2, const float* __restrict__ b2,
                         const float* __restrict__ m2, const float* __restrict__ v2,
                         float* __restrict__ p1) {
  const int TP = T_ / 8;  // 192
  int idx = blockIdx.x * 256 + threadIdx.x;
  if (idx >= B_ * F2_ * TP) return;
  int j = idx % TP;
  int c = (idx / TP) % F2_;
  int b = idx / (TP * F2_);
  float a  = g2[c] * rsqrtf(v2[c] + EPS_);
  float cc = b2[c] - a * m2[c];
  const float* src = gbuf + (b * F2_ + c) * T_ + j * 8;
  float s = 0.f;
#pragma unroll
  for (int p = 0; p < 8; ++p) {
    float u = a * src[p] + cc;
    s += u > 0.f ? u : __expf(u) - 1.f;
  }
  p1[(b * F2_ + c) * TP + j] = s * 0.125f;
}

// ---------------------------------------------------------------------------
// Kernel 3: conv3 (implicit GEMM, K=ci*k=512 via 128 WMMA k-steps)
//           + bn3 + ELU + avgpool4 -> out (B,F2,1,48)
//   Block = (b, 16-wide t2 tile), 2 waves = 2 ho-tiles.
// ---------------------------------------------------------------------------
__global__ __launch_bounds__(64) void k3_conv3(
    const float* __restrict__ p1, const float* __restrict__ w3,
    const float* __restrict__ g3, const float* __restrict__ b3,
    const float* __restrict__ m3, const float* __restrict__ v3,
    float* __restrict__ out) {
  const int TP = T_ / 8;  // 192
  __shared__ float p1s[F2_][32];  // 31 cols used: t2 window tb-7 .. tb+23
  __shared__ float ct[F2_][16];
  const int tid = threadIdx.x;
  const int b = blockIdx.y;
  const int tb = blockIdx.x * 16;

  for (int i = tid; i < F2_ * 31; i += 64) {
    int ci = i / 31, col = i % 31;
    int tg = tb - 7 + col;  // conv 'same' left pad = 7
    p1s[ci][col] = (tg >= 0 && tg < TP) ? p1[(b * F2_ + ci) * TP + tg] : 0.f;
  }
  __syncthreads();

  const int lane  = tid & 31;
  const int wv    = tid >> 5;
  const int hoOff = wv * 16;
  const int ln15  = lane & 15;
  const int lhalf = lane >> 4;

  v8f acc = {0.f, 0.f, 0.f, 0.f, 0.f, 0.f, 0.f, 0.f};
#pragma unroll 4
  for (int kb = 0; kb < (F2_ * K3_) / 4; ++kb) {  // 128 k-steps
    int kk0 = kb * 4 + 2 * lhalf;
    int ci0 = kk0 >> 4, k0 = kk0 & 15;
    int kk1 = kk0 + 1;
    int ci1 = kk1 >> 4, k1 = kk1 & 15;
    v2f a, bm;
    a.x  = w3[((hoOff + ln15) * F2_ + ci0) * K3_ + k0];  // OIHW
    a.y  = w3[((hoOff + ln15) * F2_ + ci1) * K3_ + k1];
    bm.x = p1s[ci0][ln15 + k0];
    bm.y = p1s[ci1][ln15 + k1];
    acc = wmma_f32_k4(a, bm, acc);
  }
#pragma unroll
  for (int i = 0; i < 8; ++i) {
    int ho = hoOff + i + 8 * lhalf;
    float aa = g3[ho] * rsqrtf(v3[ho] + EPS_);
    float cc = b3[ho] - aa * m3[ho];
    float u = aa * acc[i] + cc;
    ct[ho][ln15] = u > 0.f ? u : __expf(u) - 1.f;
  }
  __syncthreads();
  for (int idx = tid; idx < F2_ * 4; idx += 64) {
    int ho = idx >> 2, g = idx & 3;
    float s = (ct[ho][g * 4] + ct[ho][g * 4 + 1] + ct[ho][g * 4 + 2] +
               ct[ho][g * 4 + 3]) * 0.25f;
    out[(b * F2_ + ho) * 48 + (blockIdx.x * 4 + g)] = s;
  }
}

// ---------------------------------------------------------------------------
extern "C" void kernel_launch(void* const* d_in, const int* in_sizes, int n_in,
                              void* d_out, int out_size, void* d_ws, size_t ws_size,
                              hipStream_t stream) {
  (void)in_sizes; (void)n_in; (void)out_size; (void)ws_size;
  const float* x   = (const float*)d_in[0];
  const float* w1  = (const float*)d_in[1];
  const float* g1  = (const float*)d_in[2];
  const float* b1  = (const float*)d_in[3];
  const float* m1  = (const float*)d_in[4];
  const float* v1  = (const float*)d_in[5];
  const float* gw  = (const float*)d_in[6];
  const float* as_ = (const float*)d_in[7];
  const float* ad_ = (const float*)d_in[8];
  const float* gb  = (const float*)d_in[9];
  const float* g2  = (const float*)d_in[10];
  const float* b2  = (const float*)d_in[11];
  const float* m2  = (const float*)d_in[12];
  const float* v2  = (const float*)d_in[13];
  const float* w3  = (const float*)d_in[14];
  const float* g3  = (const float*)d_in[15];
  const float* b3  = (const float*)d_in[16];
  const float* m3  = (const float*)d_in[17];
  const float* v3  = (const float*)d_in[18];
  const int* edge  = (const int*)d_in[19];

  float* ws   = (float*)d_ws;
  float* wsW  = ws;                          // 1024 floats
  float* wsCb = ws + 1024;                   // 32 floats
  float* gbuf = ws + 2048;                   // B*F2*T = 786432 floats (3 MB)
  float* p1b  = ws + 2048 + B_ * F2_ * T_;   // B*F2*192 = 98304 floats

  k0_fold<<<1, 256, 0, stream>>>(w1, g1, b1, m1, v1, gw, wsW, wsCb);
  k1_gat<<<dim3(T_ / TT_, B_), 256, 0, stream>>>(x, as_, ad_, gb, edge, wsW,
                                                 wsCb, gbuf);
  int n2 = B_ * F2_ * (T_ / 8);
  k2_pool8<<<(n2 + 255) / 256, 256, 0, stream>>>(gbuf, g2, b2, m2, v2, p1b);
  k3_conv3<<<dim3((T_ / 8) / 16, B_), 64, 0, stream>>>(p1b, w3, g3, b3, m3, v3,
                                                       (float*)d_out);
}